// MeshHandler_69853348102547
// MI455X (gfx1250) — compile-verified
//
#include <hip/hip_runtime.h>
#include <hip/hip_bf16.h>

typedef __attribute__((ext_vector_type(2))) float v2f;
typedef __attribute__((ext_vector_type(8))) float v8f;

// sigmoid via hardware exp + hardware reciprocal (v_exp_f32 + v_rcp_f32,
// both single TRANS ops; avoids the ~12-instruction IEEE divide sequence).
__device__ __forceinline__ float sigm(float x) {
    return __builtin_amdgcn_rcpf(1.0f + __expf(-x));
}

// D = A(16x4 f32) x B(4x16 f32) + C(16x16 f32), full fp32 WMMA.
__device__ __forceinline__ v8f wmma4(v2f a, v2f b, v8f c) {
    return __builtin_amdgcn_wmma_f32_16x16x4_f32(
        /*neg_a=*/false, a, /*neg_b=*/false, b,
        /*c_mod=*/(short)0, c, /*reuse_a=*/false, /*reuse_b=*/false);
}

__global__ void zero_f32(float* __restrict__ out, int n) {
    int i = blockIdx.x * blockDim.x + threadIdx.x;
    int stride = gridDim.x * blockDim.x;
    for (; i < n; i += stride) out[i] = 0.0f;
}

__global__ __launch_bounds__(256)
void mesh_mlp_wmma(const float2* __restrict__ pts,
                   const int* __restrict__ adj,
                   const float* __restrict__ W1, const float* __restrict__ b1,
                   const float* __restrict__ W2, const float* __restrict__ b2,
                   const float* __restrict__ W3, const float* __restrict__ b3,
                   const float* __restrict__ W4, const float* __restrict__ b4,
                   float* __restrict__ out,
                   long long numTiles) {
    const int lane = threadIdx.x & 31;
    const int waveInBlock = threadIdx.x >> 5;
    const int wavesPerBlock = blockDim.x >> 5;
    long long wave = (long long)blockIdx.x * wavesPerBlock + waveInBlock;
    long long waveStride = (long long)gridDim.x * wavesPerBlock;

    const int  m    = lane & 15;          // A-matrix row (output feature)
    const int  kOff = (lane >> 4) << 1;   // lanes 16-31 carry K=2,3 of each block
    const bool low  = lane < 16;
    const int  lo   = lane & 15;
    const int  hi   = lane | 16;

    // ---- A fragments: A = W^T, padded to 16 rows / K multiple of 4 ----
    v2f A[8];
    {
        const float* Ws[4] = {W1, W2, W3, W4};
        const int Ks[4] = {6, 8, 8, 8};
        const int Ms[4] = {8, 8, 8, 3};
        #pragma unroll
        for (int l = 0; l < 4; ++l) {
            #pragma unroll
            for (int blk = 0; blk < 2; ++blk) {
                int k0 = blk * 4 + kOff;
                v2f a;
                a[0] = (m < Ms[l] && (k0    ) < Ks[l]) ? Ws[l][(k0    ) * Ms[l] + m] : 0.0f;
                a[1] = (m < Ms[l] && (k0 + 1) < Ks[l]) ? Ws[l][(k0 + 1) * Ms[l] + m] : 0.0f;
                A[l * 2 + blk] = a;
            }
        }
    }
    // ---- Bias goes in through the C accumulator (rows 0..fan_out-1) ----
    v8f Cb[4];
    {
        const float* bs[4] = {b1, b2, b3, b4};
        const int Ms[4] = {8, 8, 8, 3};
        #pragma unroll
        for (int l = 0; l < 4; ++l) {
            #pragma unroll
            for (int r = 0; r < 8; ++r)
                Cb[l][r] = (low && r < Ms[l]) ? bs[l][r] : 0.0f;
        }
    }

    // Run one 16-element MLP chain. b00/b01 are the layer-1 B fragments
    // (K-blocks 0 and 1). Returns pre-sigmoid layer-4 D (rows 0..2 valid,
    // columns = lanes 0-15).
    auto run_chain = [&](v2f b00, v2f b01) -> v8f {
        v8f d = wmma4(A[1], b01, Cb[0]);
        d = wmma4(A[0], b00, d);
        #pragma unroll
        for (int l = 1; l < 4; ++l) {
            float h[8];
            #pragma unroll
            for (int r = 0; r < 8; ++r) h[r] = sigm(d[r]);
            // lanes 16-31 need features 2,3 / 6,7 of "their" column (in lane lo)
            float t2 = __shfl(h[2], lo, 32);
            float t3 = __shfl(h[3], lo, 32);
            float t6 = __shfl(h[6], lo, 32);
            float t7 = __shfl(h[7], lo, 32);
            v2f nb0, nb1;
            nb0[0] = low ? h[0] : t2;  nb0[1] = low ? h[1] : t3;
            nb1[0] = low ? h[4] : t6;  nb1[1] = low ? h[5] : t7;
            d = wmma4(A[l * 2 + 1], nb1, Cb[l]);
            d = wmma4(A[l * 2    ], nb0, d);
        }
        return d;
    };

    for (long long tile = wave; tile < numTiles; tile += waveStride) {
        long long e  = tile * 32 + lane;     // this lane's element
        long long a3 = e * 3;
        int i0 = adj[a3 + 0];
        int i1 = adj[a3 + 1];
        int i2 = adj[a3 + 2];
        float2 q0 = pts[i0];
        float2 q1 = pts[i1];
        float2 q2 = pts[i2];

        // Chain 0: elements in lanes 0-15.  B col n = elem n.
        // K=0..3 -> (x0,y0,x1,y1); K=4..7 -> (x2,y2,0,0)
        float sx1 = __shfl(q1.x, lo, 32);
        float sy1 = __shfl(q1.y, lo, 32);
        v2f c0b0, c0b1;
        c0b0[0] = low ? q0.x : sx1;  c0b0[1] = low ? q0.y : sy1;
        c0b1[0] = low ? q2.x : 0.f;  c0b1[1] = low ? q2.y : 0.f;

        // Chain 1: elements in lanes 16-31.  B col n = elem 16+n.
        float hx0 = __shfl(q0.x, hi, 32);
        float hy0 = __shfl(q0.y, hi, 32);
        float hx2 = __shfl(q2.x, hi, 32);
        float hy2 = __shfl(q2.y, hi, 32);
        v2f c1b0, c1b1;
        c1b0[0] = low ? hx0 : q1.x;  c1b0[1] = low ? hy0 : q1.y;
        c1b1[0] = low ? hx2 : 0.f;   c1b1[1] = low ? hy2 : 0.f;

        v8f d0 = run_chain(c0b0, c0b1);
        v8f d1 = run_chain(c1b0, c1b1);

        // Final sigmoid + route chain-1 weights up to lanes 16-31.
        float w[3];
        #pragma unroll
        for (int v = 0; v < 3; ++v) {
            float w0 = sigm(d0[v]);
            float w1 = sigm(d1[v]);
            float up = __shfl(w1, lo, 32);
            w[v] = low ? w0 : up;
        }
        atomicAdd(out + i0, w[0]);
        atomicAdd(out + i1, w[1]);
        atomicAdd(out + i2, w[2]);
    }
}

// Scalar fallback for a non-multiple-of-32 tail (not hit at E = 4M).
__global__ void mesh_mlp_tail(const float2* __restrict__ pts,
                              const int* __restrict__ adj,
                              const float* __restrict__ W1, const float* __restrict__ b1,
                              const float* __restrict__ W2, const float* __restrict__ b2,
                              const float* __restrict__ W3, const float* __restrict__ b3,
                              const float* __restrict__ W4, const float* __restrict__ b4,
                              float* __restrict__ out,
                              long long start, long long E) {
    long long e = start + (long long)blockIdx.x * blockDim.x + threadIdx.x;
    if (e >= E) return;
    long long a3 = e * 3;
    int idx[3] = {adj[a3], adj[a3 + 1], adj[a3 + 2]};
    float x[6];
    #pragma unroll
    for (int v = 0; v < 3; ++v) {
        float2 q = pts[idx[v]];
        x[2 * v] = q.x; x[2 * v + 1] = q.y;
    }
    float h1[8], h2[8], h3[8];
    #pragma unroll
    for (int j = 0; j < 8; ++j) {
        float s = b1[j];
        #pragma unroll
        for (int k = 0; k < 6; ++k) s += x[k] * W1[k * 8 + j];
        h1[j] = sigm(s);
    }
    #pragma unroll
    for (int j = 0; j < 8; ++j) {
        float s = b2[j];
        #pragma unroll
        for (int k = 0; k < 8; ++k) s += h1[k] * W2[k * 8 + j];
        h2[j] = sigm(s);
    }
    #pragma unroll
    for (int j = 0; j < 8; ++j) {
        float s = b3[j];
        #pragma unroll
        for (int k = 0; k < 8; ++k) s += h2[k] * W3[k * 8 + j];
        h3[j] = sigm(s);
    }
    #pragma unroll
    for (int v = 0; v < 3; ++v) {
        float s = b4[v];
        #pragma unroll
        for (int k = 0; k < 8; ++k) s += h3[k] * W4[k * 3 + v];
        atomicAdd(out + idx[v], sigm(s));
    }
}

extern "C" void kernel_launch(void* const* d_in, const int* in_sizes, int n_in,
                              void* d_out, int out_size, void* d_ws, size_t ws_size,
                              hipStream_t stream) {
    const float2* pts = (const float2*)d_in[0];
    const int*    adj = (const int*)d_in[1];
    const float*  W1 = (const float*)d_in[2];
    const float*  b1 = (const float*)d_in[3];
    const float*  W2 = (const float*)d_in[4];
    const float*  b2 = (const float*)d_in[5];
    const float*  W3 = (const float*)d_in[6];
    const float*  b3 = (const float*)d_in[7];
    const float*  W4 = (const float*)d_in[8];
    const float*  b4 = (const float*)d_in[9];
    float* out = (float*)d_out;

    long long E = (long long)in_sizes[1] / 3;

    // 1) Zero the accumulator (harness poisons d_out; atomics accumulate).
    {
        int blocks = (out_size + 255) / 256;
        if (blocks > 8192) blocks = 8192;
        if (blocks < 1) blocks = 1;
        zero_f32<<<blocks, 256, 0, stream>>>(out, out_size);
    }

    // 2) Main WMMA path: 32 elements per wave, 8 waves per block.
    long long tiles = E / 32;
    if (tiles > 0) {
        const int wavesPerBlock = 8;
        long long blocksNeeded = (tiles + wavesPerBlock - 1) / wavesPerBlock;
        int blocks = blocksNeeded > 8192 ? 8192 : (int)blocksNeeded;
        mesh_mlp_wmma<<<blocks, 256, 0, stream>>>(
            pts, adj, W1, b1, W2, b2, W3, b3, W4, b4, out, tiles);
    }

    // 3) Scalar tail (no-op for E % 32 == 0).
    long long rem = E - tiles * 32;
    if (rem > 0) {
        int tb = 64;
        int tg = (int)((rem + tb - 1) / tb);
        mesh_mlp_tail<<<tg, tb, 0, stream>>>(
            pts, adj, W1, b1, W2, b2, W3, b3, W4, b4, out, tiles * 32, E);
    }
}